// NpiLoss_12652973654366
// MI455X (gfx1250) — compile-verified
//
#include <hip/hip_runtime.h>

// NPI epidemic-renewal loss for gfx1250 (MI455X, wave32).
//
// Phase 1 (cases): blocked triangular recurrence. History contribution of each
//   16-row time block is a GEMM with region-shared weights W[r][j] = SI[i-j],
//   computed with V_WMMA_F32_16X16X4_F32 (f32 precision = reference).
//   The 16x16 in-block triangular tail is sequential per lane (1 lane = 1 region).
// Phase 2 (deaths): per-region kernel f[:,m] -> per-lane convolution against the
//   pred column kept in LDS, 16 output rows per chunk so each f[d,m] load feeds
//   16 FMAs.
// Reduction: deterministic (fixed-order sums, no float atomics).

typedef float v2f __attribute__((ext_vector_type(2)));
typedef float v8f __attribute__((ext_vector_type(8)));

#define N2        128
#define MREG      16384
#define WGCOLS    64           // columns (regions) per workgroup, 2 waves
#define PSTRIDE   72           // pred LDS row stride (floats): 2 rows apart -> banks +16
#define TSTRIDE   68           // tconv LDS row stride: 8 rows apart -> banks +32

__global__ __launch_bounds__(WGCOLS) void npi_loss_kernel(
    const float* __restrict__ rt, const float* __restrict__ deaths_gt,
    const float* __restrict__ SI, const float* __restrict__ f,
    const float* __restrict__ seed, const int* __restrict__ start_p,
    float* __restrict__ partials)
{
    __shared__ float pred_lds[N2 * PSTRIDE];   // 36 KB
    __shared__ float tconv[16 * TSTRIDE];      // 4.25 KB  (per-wave column-partitioned)
    __shared__ float si_lds[N2];
    __shared__ float red[WGCOLS];

    const int tid  = threadIdx.x;
    const int lane = tid & 31;
    const int wave = tid >> 5;
    const int cc   = tid;                         // column within LDS tile
    const int m    = blockIdx.x * WGCOLS + tid;   // global region index
    const int start = *start_p;                   // == 6

    // Stage SI into LDS (needed lane-divergently for WMMA A-fragments).
    si_lds[tid]      = SI[tid];
    si_lds[tid + 64] = SI[tid + 64];
    __syncthreads();

    // Small lags (1..15) for the in-block triangular tail: uniform scalar loads.
    float sreg[16];
    sreg[0] = 0.0f;
#pragma unroll
    for (int d = 1; d < 16; ++d) sreg[d] = SI[d];

    const int half = lane >> 4;       // lane half selects K {0,1} vs {2,3} / rows q vs q+8
    const int lm   = lane & 15;
    const int nt0  = wave * 32;       // this wave's two 16-wide N-tiles
    const int nt1  = wave * 32 + 16;

    // ---------------- Phase 1: predict cases ----------------
#pragma unroll
    for (int b = 0; b < 8; ++b) {
        const int i0 = b * 16;
        float hist[16];
        if (b == 0) {
#pragma unroll
            for (int r = 0; r < 16; ++r) hist[r] = 0.0f;
        } else {
            // History GEMM: T[16][32] = W[16][i0] * pred[0:i0][32] for this wave.
            v8f c0 = {0,0,0,0,0,0,0,0};
            v8f c1 = {0,0,0,0,0,0,0,0};
            const int irow = i0 + lm;             // A-fragment output row (time i)
            for (int j0 = 0; j0 < i0; j0 += 4) {
                const int ja = j0 + 2 * half;     // K index for this lane half
                v2f a;                            // A[16x4] f32: W[r][j] = SI[i - j]
                a.x = si_lds[irow - ja];          // i - j >= 1 always
                a.y = si_lds[irow - ja - 1];
                v2f b0, b1;                       // B[4x16] f32 from pred LDS
                b0.x = pred_lds[ ja      * PSTRIDE + nt0 + lm];
                b0.y = pred_lds[(ja + 1) * PSTRIDE + nt0 + lm];
                b1.x = pred_lds[ ja      * PSTRIDE + nt1 + lm];
                b1.y = pred_lds[(ja + 1) * PSTRIDE + nt1 + lm];
                c0 = __builtin_amdgcn_wmma_f32_16x16x4_f32(
                        false, a, false, b0, (short)0, c0, false, false);
                c1 = __builtin_amdgcn_wmma_f32_16x16x4_f32(
                        false, a, false, b1, (short)0, c1, false, false);
            }
            // Scatter C fragments (row = q + 8*half, col = tile + lm) to LDS,
            // then gather each lane's own column. Same-wave LDS is in-order:
            // no barrier needed (columns are wave-partitioned).
#pragma unroll
            for (int q = 0; q < 8; ++q) {
                tconv[(q + 8 * half) * TSTRIDE + nt0 + lm] = c0[q];
                tconv[(q + 8 * half) * TSTRIDE + nt1 + lm] = c1[q];
            }
#pragma unroll
            for (int r = 0; r < 16; ++r) hist[r] = tconv[r * TSTRIDE + cc];
        }

        // Prefetch rt rows for this block (coalesced, hides latency before the chain).
        float rt_blk[16];
#pragma unroll
        for (int r = 0; r < 16; ++r) rt_blk[r] = rt[(i0 + r) * MREG + m];

        // Sequential in-block triangular recurrence (registers only).
        float pblk[16];
#pragma unroll
        for (int r = 0; r < 16; ++r) {
            const int i = i0 + r;
            float conv = hist[r];
#pragma unroll
            for (int r2 = 0; r2 < r; ++r2) conv += sreg[r - r2] * pblk[r2];
            float val;
            if (i < start) {                      // uniform branch; only block 0
                val = seed[i * MREG + m];
            } else {
                val = rt_blk[r] * conv;
            }
            pblk[r] = val;
            pred_lds[i * PSTRIDE + cc] = val;
        }
    }

    // ---------------- Phase 2: deaths convolution + squared error ----------------
    // E[i] = sum_{d=2..i} pred[i-d] * f[d,m];  rows i<start overridden (1e-9 at i=0).
    float err = 0.0f;
    for (int i0 = 0; i0 < N2; i0 += 16) {
        float e[16];
#pragma unroll
        for (int r = 0; r < 16; ++r) e[r] = 0.0f;
        // Full part: d = 2..i0 hits all 16 output rows.
        for (int d = 2; d <= i0; ++d) {
            const float fd = f[d * MREG + m];
#pragma unroll
            for (int r = 0; r < 16; ++r)
                e[r] += pred_lds[(i0 + r - d) * PSTRIDE + cc] * fd;
        }
        // Triangular tail: d = i0+dr hits rows r >= dr, pred index = r - dr.
#pragma unroll
        for (int dr = 1; dr < 16; ++dr) {
            const int d = i0 + dr;
            if (d >= 2) {
                const float fd = f[d * MREG + m];
#pragma unroll
                for (int r = dr; r < 16; ++r)
                    e[r] += pred_lds[(r - dr) * PSTRIDE + cc] * fd;
            }
        }
#pragma unroll
        for (int r = 0; r < 16; ++r) {
            const int i = i0 + r;
            float Ei = e[r];
            if (i < start) Ei = (i == 0) ? 1e-9f : 0.0f;
            const float diff = Ei - deaths_gt[i * MREG + m];
            err += diff * diff;
        }
    }

    // ---------------- Deterministic workgroup reduction ----------------
    red[tid] = err;
    __syncthreads();
    if (tid == 0) {
        float s = 0.0f;
        for (int t = 0; t < WGCOLS; ++t) s += red[t];
        partials[blockIdx.x] = s;
    }
}

__global__ void npi_finalize(const float* __restrict__ partials,
                             float* __restrict__ out, int nblocks)
{
    if (blockIdx.x == 0 && threadIdx.x == 0) {
        float s = 0.0f;
        for (int i = 0; i < nblocks; ++i) s += partials[i];
        out[0] = s * (1.0f / ((float)N2 * (float)MREG));
    }
}

extern "C" void kernel_launch(void* const* d_in, const int* in_sizes, int n_in,
                              void* d_out, int out_size, void* d_ws, size_t ws_size,
                              hipStream_t stream) {
    (void)in_sizes; (void)n_in; (void)out_size; (void)ws_size;
    const float* rt    = (const float*)d_in[0];
    const float* gt    = (const float*)d_in[1];
    const float* SI    = (const float*)d_in[2];
    const float* f     = (const float*)d_in[3];
    const float* seed  = (const float*)d_in[4];
    const int*   start = (const int*)d_in[5];
    float* out      = (float*)d_out;
    float* partials = (float*)d_ws;               // 256 floats of scratch

    const int nblocks = MREG / WGCOLS;            // 256 workgroups
    npi_loss_kernel<<<nblocks, WGCOLS, 0, stream>>>(rt, gt, SI, f, seed, start, partials);
    npi_finalize<<<1, 32, 0, stream>>>(partials, out, nblocks);
}